// RadialL2Loss_24515673326163
// MI455X (gfx1250) — compile-verified
//
#include <hip/hip_runtime.h>

// ---------------------------------------------------------------------------
// RadialL2Loss on MI455X (gfx1250).
// HBM-bound streaming reduction: 20 B/point * 2^24 points = 335 MB -> ~14.4 us
// floor at 23.3 TB/s. Data path: double-buffered async global->LDS copies
// (global_load_async_to_lds_b128, ASYNCcnt) with lane-private LDS slots,
// consumed via ds_load_b128. The async 'offset:' immediate is added to BOTH
// the global and LDS addresses (ISA 08 4.4), so each lane's 32B slab is two
// b128 ops sharing one address pair. Binning uses suffix-sum accumulators
// (S_k over dist > p_k): 1 add + 3 fma per accumulated quantity; exact
// boundary points (dist == 0.25/0.5/0.75, inclusive in BOTH bins per the
// reference) are re-accumulated once with >= masks under a rare branch.
// Deterministic 2-stage reduction (no float atomics).
// ---------------------------------------------------------------------------

#define TPB 256
#define BLOCKS 1024
#define PTS_PER_LANE 4
#define TILE_PTS (TPB * PTS_PER_LANE) // 1024 points per block per tile

// LDS layout (single __shared__ object => base offset 0).
#define XBUF_OFF 0u      // 2 bufs * 256 lanes * 32 B = 16384
#define YBUF_OFF 16384u  // 16384
#define DBUF_OFF 32768u  // 2 bufs * 256 lanes * 16 B = 8192
#define RED_OFF 40960u   // 8 waves * 12 floats = 384
#define SMEM_BYTES (40960u + 512u)

__device__ __forceinline__ void accum_point(float x0, float x1, float y0, float y1,
                                            float d, float sd[4], float sa[4],
                                            float sc[4]) {
  float dx0 = x0 - y0, dx1 = x1 - y1;
  float d2 = fmaf(dx1, dx1, dx0 * dx0);
  float nx = fmaf(x1, x1, x0 * x0);
  float ny = fmaf(y1, y1, y0 * y0);
  float t = __builtin_amdgcn_sqrtf(nx) - __builtin_amdgcn_sqrtf(ny);
  float a = t * t;
  float g1 = (d > 0.25f) ? 1.0f : 0.0f;
  float g2 = (d > 0.50f) ? 1.0f : 0.0f;
  float g3 = (d > 0.75f) ? 1.0f : 0.0f;
  sd[0] += d2;
  sd[1] = fmaf(g1, d2, sd[1]);
  sd[2] = fmaf(g2, d2, sd[2]);
  sd[3] = fmaf(g3, d2, sd[3]);
  sa[0] += a;
  sa[1] = fmaf(g1, a, sa[1]);
  sa[2] = fmaf(g2, a, sa[2]);
  sa[3] = fmaf(g3, a, sa[3]);
  sc[0] += 1.0f;
  sc[1] += g1;
  sc[2] += g2;
  sc[3] += g3;
  // Exact boundary hit: reference masks are inclusive on both ends, so the
  // point belongs to two bins. Re-accumulate once with >= masks (rare branch).
  if (__builtin_expect((d == 0.25f) || (d == 0.50f) || (d == 0.75f), 0)) {
    float e1 = (d >= 0.25f) ? 1.0f : 0.0f;
    float e2 = (d >= 0.50f) ? 1.0f : 0.0f;
    float e3 = (d >= 0.75f) ? 1.0f : 0.0f;
    sd[0] += d2;
    sd[1] = fmaf(e1, d2, sd[1]);
    sd[2] = fmaf(e2, d2, sd[2]);
    sd[3] = fmaf(e3, d2, sd[3]);
    sa[0] += a;
    sa[1] = fmaf(e1, a, sa[1]);
    sa[2] = fmaf(e2, a, sa[2]);
    sa[3] = fmaf(e3, a, sa[3]);
    sc[0] += 1.0f;
    sc[1] += e1;
    sc[2] += e2;
    sc[3] += e3;
  }
}

__global__ __launch_bounds__(TPB) void radial_stage1(const float* __restrict__ x,
                                                     const float* __restrict__ y,
                                                     const float* __restrict__ dist,
                                                     float* __restrict__ partial,
                                                     int numTiles, int N) {
  __shared__ __align__(16) unsigned char smem[SMEM_BYTES];
  const int tid = threadIdx.x;
  const unsigned lane32 = (unsigned)tid * 32u;
  const unsigned lane16 = (unsigned)tid * 16u;

  float sd[4] = {0.f, 0.f, 0.f, 0.f};
  float sa[4] = {0.f, 0.f, 0.f, 0.f};
  float sc[4] = {0.f, 0.f, 0.f, 0.f};

  // Lane-private async copies: x quad (32B = 2*b128), y quad (32B), dist quad
  // (16B). 'offset:16' shifts BOTH global and LDS addresses, reusing the same
  // address registers for each lane's second 16B chunk.
  auto issue_tile = [&](int tile, int b) {
    long long p0 = (long long)tile * TILE_PTS + (long long)tid * PTS_PER_LANE;
    const char* gx = (const char*)x + p0 * 8; // 2 floats per point
    const char* gy = (const char*)y + p0 * 8;
    const char* gd = (const char*)dist + p0 * 4;
    unsigned ox = XBUF_OFF + (unsigned)b * 8192u + lane32;
    unsigned oy = YBUF_OFF + (unsigned)b * 8192u + lane32;
    unsigned od = DBUF_OFF + (unsigned)b * 4096u + lane16;
    asm volatile("global_load_async_to_lds_b128 %0, %1, off" ::"v"(ox), "v"(gx)
                 : "memory");
    asm volatile("global_load_async_to_lds_b128 %0, %1, off offset:16" ::"v"(ox),
                 "v"(gx)
                 : "memory");
    asm volatile("global_load_async_to_lds_b128 %0, %1, off" ::"v"(oy), "v"(gy)
                 : "memory");
    asm volatile("global_load_async_to_lds_b128 %0, %1, off offset:16" ::"v"(oy),
                 "v"(gy)
                 : "memory");
    asm volatile("global_load_async_to_lds_b128 %0, %1, off" ::"v"(od), "v"(gd)
                 : "memory");
  };

  int t = blockIdx.x;
  int buf = 0;
  if (t < numTiles) issue_tile(t, 0);
  for (; t < numTiles; t += gridDim.x) {
    int nxt = t + gridDim.x;
    if (nxt < numTiles) {
      // Buffer buf^1 was consumed last iteration; make sure its ds reads
      // retired before the async engine overwrites it.
      asm volatile("s_wait_dscnt 0x0" ::: "memory");
      issue_tile(nxt, buf ^ 1);
      asm volatile("s_wait_asynccnt 0x5" ::: "memory"); // current tile's 5 done
    } else {
      asm volatile("s_wait_asynccnt 0x0" ::: "memory"); // drain
    }
    const unsigned xo = XBUF_OFF + (unsigned)buf * 8192u + lane32;
    const unsigned yo = YBUF_OFF + (unsigned)buf * 8192u + lane32;
    const unsigned dz = DBUF_OFF + (unsigned)buf * 4096u + lane16;
    const float4 xv0 = *(const float4*)(smem + xo);
    const float4 xv1 = *(const float4*)(smem + xo + 16u);
    const float4 yv0 = *(const float4*)(smem + yo);
    const float4 yv1 = *(const float4*)(smem + yo + 16u);
    const float4 dv = *(const float4*)(smem + dz);
    accum_point(xv0.x, xv0.y, yv0.x, yv0.y, dv.x, sd, sa, sc);
    accum_point(xv0.z, xv0.w, yv0.z, yv0.w, dv.y, sd, sa, sc);
    accum_point(xv1.x, xv1.y, yv1.x, yv1.y, dv.z, sd, sa, sc);
    accum_point(xv1.z, xv1.w, yv1.z, yv1.w, dv.w, sd, sa, sc);
    buf ^= 1;
  }

  // Tail guard (empty for N = 2^24): plain global loads from block 0.
  if (blockIdx.x == 0) {
    for (int p = numTiles * TILE_PTS + tid; p < N; p += TPB) {
      float2 xv = ((const float2*)x)[p];
      float2 yv = ((const float2*)y)[p];
      accum_point(xv.x, xv.y, yv.x, yv.y, dist[p], sd, sa, sc);
    }
  }

  // Deterministic block reduction: wave32 shuffle tree, then cross-wave in LDS.
  float v[12];
#pragma unroll
  for (int i = 0; i < 4; ++i) {
    v[i] = sd[i];
    v[4 + i] = sa[i];
    v[8 + i] = sc[i];
  }
#pragma unroll
  for (int off = 16; off >= 1; off >>= 1) {
#pragma unroll
    for (int i = 0; i < 12; ++i) v[i] += __shfl_xor(v[i], off, 32);
  }
  float* red = (float*)(smem + RED_OFF);
  int wave = tid >> 5, lane = tid & 31;
  if (lane == 0) {
#pragma unroll
    for (int i = 0; i < 12; ++i) red[wave * 12 + i] = v[i];
  }
  __syncthreads();
  if (tid < 12) {
    float s = 0.f;
#pragma unroll
    for (int w = 0; w < TPB / 32; ++w) s += red[w * 12 + tid];
    partial[blockIdx.x * 12 + tid] = s;
  }
}

__global__ __launch_bounds__(TPB) void radial_stage2(const float* __restrict__ partial,
                                                     int rows, float* __restrict__ out) {
  double acc[12];
#pragma unroll
  for (int i = 0; i < 12; ++i) acc[i] = 0.0;
  for (int r = threadIdx.x; r < rows; r += blockDim.x) {
#pragma unroll
    for (int i = 0; i < 12; ++i) acc[i] += (double)partial[r * 12 + i];
  }
#pragma unroll
  for (int off = 16; off >= 1; off >>= 1) {
#pragma unroll
    for (int i = 0; i < 12; ++i) acc[i] += __shfl_xor(acc[i], off, 32);
  }
  __shared__ double red[(TPB / 32) * 12];
  int wave = threadIdx.x >> 5, lane = threadIdx.x & 31;
  if (lane == 0) {
#pragma unroll
    for (int i = 0; i < 12; ++i) red[wave * 12 + i] = acc[i];
  }
  __syncthreads();
  if (threadIdx.x == 0) {
    double D[5], A[5], C[5];
    for (int i = 0; i < 4; ++i) {
      double sD = 0.0, sA = 0.0, sC = 0.0;
      for (int w = 0; w < TPB / 32; ++w) {
        sD += red[w * 12 + i];
        sA += red[w * 12 + 4 + i];
        sC += red[w * 12 + 8 + i];
      }
      D[i] = sD;
      A[i] = sA;
      C[i] = sC;
    }
    D[4] = A[4] = C[4] = 0.0;
    double loss = 0.0;
    for (int i = 0; i < 4; ++i) {
      double bd = D[i] - D[i + 1];
      double ba = A[i] - A[i + 1];
      double bc = C[i] - C[i + 1];
      if (bc > 0.0) loss += bd / (2.0 * bc) + 0.1 * ba / bc;
    }
    out[0] = (float)loss;
  }
}

extern "C" void kernel_launch(void* const* d_in, const int* in_sizes, int n_in,
                              void* d_out, int out_size, void* d_ws, size_t ws_size,
                              hipStream_t stream) {
  const float* x = (const float*)d_in[0];    // [N,2] f32
  const float* y = (const float*)d_in[1];    // [N,2] f32
  const float* dist = (const float*)d_in[2]; // [N]   f32
  float* partial = (float*)d_ws;             // BLOCKS * 12 floats = 48 KB
  const int N = in_sizes[2];
  const int numTiles = N / TILE_PTS; // N = 2^24 -> 16384 tiles, exact
  radial_stage1<<<BLOCKS, TPB, 0, stream>>>(x, y, dist, partial, numTiles, N);
  radial_stage2<<<1, TPB, 0, stream>>>(partial, BLOCKS, (float*)d_out);
}